// GCNModel_27410481283793
// MI455X (gfx1250) — compile-verified
//
#include <hip/hip_runtime.h>
#include <hip/hip_bf16.h>
#include <math.h>

typedef float v2f __attribute__((ext_vector_type(2)));
typedef float v8f __attribute__((ext_vector_type(8)));

#define HID 64

// ---------------------------------------------------------------------------
// out[rows x 64] = A[rows x K] @ W[K x 64], fp32 WMMA 16x16x4.
// If ids != nullptr, row m of A is A[ids[m]*K + k] (fused embedding gather).
// blockDim = 128 (4 waves); wave w computes the 16x16 tile at cols [16w,16w+16).
// gridDim.x = rows/16 (rows % 16 == 0 guaranteed: N = 100000).
// No divergence anywhere -> EXEC all ones for WMMA as required.
// ---------------------------------------------------------------------------
template <int K>
__global__ void wmma_gemm_f32(const float* __restrict__ A,
                              const int* __restrict__ ids,
                              const float* __restrict__ W,
                              float* __restrict__ out) {
  const int lane = threadIdx.x & 31;
  const int wave = threadIdx.x >> 5;          // 0..3 -> N tile
  const int rowBase = blockIdx.x * 16;
  const int m    = lane & 15;                 // A row within tile
  const int hi   = lane >> 4;                 // 0 or 1
  const int koff = hi << 1;                   // K sub-offset: 0 or 2
  const int col  = (wave << 4) + (lane & 15); // output column 0..63

  int arow = rowBase + m;
  if (ids) arow = ids[arow];
  const float* __restrict__ Arow = A + (long)arow * K;

  v8f acc = {};
#pragma unroll
  for (int k0 = 0; k0 < K; k0 += 4) {
    // A 16x4 fp32 operand: lane holds A[m][k0+koff], A[m][k0+koff+1]
    v2f a = *(const v2f*)(Arow + k0 + koff);   // 8B aligned (koff even, K even)
    // B 4x16 fp32 operand: lane holds W[k0+koff][col], W[k0+koff+1][col]
    v2f b;
    b.x = W[(k0 + koff) * HID + col];
    b.y = W[(k0 + koff + 1) * HID + col];
    acc = __builtin_amdgcn_wmma_f32_16x16x4_f32(
        /*neg_a=*/false, a, /*neg_b=*/false, b,
        /*c_mod=*/(short)0, acc, /*reuse_a=*/false, /*reuse_b=*/false);
  }

  // C/D layout: VGPR r -> row (r + 8*hi), col = lane&15 within tile
#pragma unroll
  for (int r = 0; r < 8; ++r) {
    out[(long)(rowBase + r + 8 * hi) * HID + col] = acc[r];
  }
}

// ---------------------------------------------------------------------------
// vectorized zero: n must be a multiple of 4 (true for all call sites)
__global__ void zero_f32x4(float4* __restrict__ p, long n4) {
  long i = (long)blockIdx.x * blockDim.x + threadIdx.x;
  if (i < n4) p[i] = make_float4(0.f, 0.f, 0.f, 0.f);
}

// deg[dst] += 1 per edge (into buffer later transformed to dinv)
__global__ void deg_count(const int* __restrict__ dst, float* __restrict__ deg, int E) {
  int e = blockIdx.x * blockDim.x + threadIdx.x;
  if (e < E) atomicAdd(&deg[dst[e]], 1.0f);
}

// dinv[i] = rsqrt(deg[i] + 1)   (self loop)
__global__ void deg_to_dinv(float* __restrict__ d, int n) {
  int i = blockIdx.x * blockDim.x + threadIdx.x;
  if (i < n) d[i] = rsqrtf(d[i] + 1.0f);
}

// agg[dst][hg..hg+3] += xw[src][hg..hg+3] * dinv[src]*dinv[dst]
// thread per (edge, 4-float group): 16 threads/edge, b128 xw read, 4 atomics.
__global__ void edge_scatter4(const int* __restrict__ src, const int* __restrict__ dst,
                              const float* __restrict__ xw, const float* __restrict__ dinv,
                              float* __restrict__ agg, int E) {
  int idx = blockIdx.x * blockDim.x + threadIdx.x;
  int e  = idx >> 4;
  int hg = (idx & 15) << 2;
  if (e >= E) return;
  int s = src[e], d = dst[e];
  float coef = dinv[s] * dinv[d];
  float4 x = *(const float4*)(xw + (long)s * HID + hg);
  float* a = agg + (long)d * HID + hg;
  atomicAdd(a + 0, x.x * coef);
  atomicAdd(a + 1, x.y * coef);
  atomicAdd(a + 2, x.z * coef);
  atomicAdd(a + 3, x.w * coef);
}

// in-place: agg = relu(agg + xw * dinv^2 + b)   (thread per (node, 4-float group))
__global__ void self_bias_relu4(float* __restrict__ agg, const float* __restrict__ xw,
                                const float* __restrict__ dinv, const float* __restrict__ b,
                                int N) {
  long idx = (long)blockIdx.x * blockDim.x + threadIdx.x;   // over N*16 groups
  if (idx >= (long)N * (HID / 4)) return;
  int i  = (int)(idx >> 4);
  int hg = (int)(idx & 15) << 2;
  float dv = dinv[i];
  float d2 = dv * dv;
  float4 a  = *(const float4*)(agg + (long)i * HID + hg);
  float4 x  = *(const float4*)(xw + (long)i * HID + hg);
  float4 bb = *(const float4*)(b + hg);
  float4 r;
  r.x = fmaxf(a.x + x.x * d2 + bb.x, 0.f);
  r.y = fmaxf(a.y + x.y * d2 + bb.y, 0.f);
  r.z = fmaxf(a.z + x.z * d2 + bb.z, 0.f);
  r.w = fmaxf(a.w + x.w * d2 + bb.w, 0.f);
  *(float4*)(agg + (long)i * HID + hg) = r;
}

// counts[batch[i]] += 1
__global__ void pool_count(const int* __restrict__ batch, float* __restrict__ counts, int N) {
  int i = blockIdx.x * blockDim.x + threadIdx.x;
  if (i < N) atomicAdd(&counts[batch[i]], 1.0f);
}

// pooled[batch[i]][hg..hg+3] += x[i][hg..hg+3]
__global__ void pool_sum4(const float* __restrict__ x, const int* __restrict__ batch,
                          float* __restrict__ pooled, int N) {
  long idx = (long)blockIdx.x * blockDim.x + threadIdx.x;   // over N*16 groups
  if (idx >= (long)N * (HID / 4)) return;
  int i  = (int)(idx >> 4);
  int hg = (int)(idx & 15) << 2;
  float4 v = *(const float4*)(x + (long)i * HID + hg);
  float* p = pooled + (long)batch[i] * HID + hg;
  atomicAdd(p + 0, v.x);
  atomicAdd(p + 1, v.y);
  atomicAdd(p + 2, v.z);
  atomicAdd(p + 3, v.w);
}

// pooled[g][h] /= max(counts[g], 1)
__global__ void pool_div(float* __restrict__ pooled, const float* __restrict__ counts, int G) {
  int idx = blockIdx.x * blockDim.x + threadIdx.x;
  if (idx >= G * HID) return;
  int g = idx >> 6;
  pooled[idx] /= fmaxf(counts[g], 1.0f);
}

// per-graph head: h = relu(pooled @ Wf1 + bf1); out = sigmoid(h @ Wf2 + bf2)
// one block (64 threads) per graph
__global__ void mlp_head(const float* __restrict__ pooled,
                         const float* __restrict__ Wf1, const float* __restrict__ bf1,
                         const float* __restrict__ Wf2, const float* __restrict__ bf2,
                         float* __restrict__ out) {
  __shared__ float p[HID];
  __shared__ float hbuf[HID];
  int g = blockIdx.x, t = threadIdx.x;
  p[t] = pooled[g * HID + t];
  __syncthreads();
  float acc = bf1[t];
  for (int k = 0; k < HID; ++k) acc += p[k] * Wf1[k * HID + t];
  hbuf[t] = fmaxf(acc, 0.0f) * Wf2[t];   // fold the [64,1] Wf2 in here
  __syncthreads();
  if (t == 0) {
    float s = bf2[0];
    for (int k = 0; k < HID; ++k) s += hbuf[k];
    out[g] = 1.0f / (1.0f + expf(-s));
  }
}

// ---------------------------------------------------------------------------
static inline size_t align256(size_t x) { return (x + 255) & ~(size_t)255; }

extern "C" void kernel_launch(void* const* d_in, const int* in_sizes, int n_in,
                              void* d_out, int out_size, void* d_ws, size_t ws_size,
                              hipStream_t stream) {
  const int*   x_ids = (const int*)d_in[0];
  const int*   eidx  = (const int*)d_in[1];
  const int*   batch = (const int*)d_in[2];
  const float* emb   = (const float*)d_in[3];
  const float* W1 = (const float*)d_in[4];  const float* b1 = (const float*)d_in[5];
  const float* W2 = (const float*)d_in[6];  const float* b2 = (const float*)d_in[7];
  const float* W3 = (const float*)d_in[8];  const float* b3 = (const float*)d_in[9];
  const float* Wf1 = (const float*)d_in[10]; const float* bf1 = (const float*)d_in[11];
  const float* Wf2 = (const float*)d_in[12]; const float* bf2 = (const float*)d_in[13];
  float* out = (float*)d_out;

  const int N = in_sizes[0];           // 100000  (divisible by 16)
  const int E = in_sizes[1] / 2;       // 1600000
  const int G = out_size;              // 256
  const int* src = eidx;               // edge_index[0]
  const int* dst = eidx + E;           // edge_index[1]

  // workspace layout
  char* ws = (char*)d_ws;
  const size_t featBytes = (size_t)N * HID * sizeof(float);
  float* bufA   = (float*)ws;                                    // features / agg
  float* bufB   = (float*)(ws + align256(featBytes));            // xw
  float* dinv   = (float*)(ws + 2 * align256(featBytes));
  float* pooled = (float*)((char*)dinv + align256((size_t)N * sizeof(float)));
  float* counts = (float*)((char*)pooled + align256((size_t)G * HID * sizeof(float)));

  const int T = 256;
  const long NH   = (long)N * HID;
  const long NH4  = NH / 4;                       // N*16, exact
  const long EH4  = (long)E * (HID / 4);          // E*16, exact
  const int gridNH4  = (int)((NH4 + T - 1) / T);
  const int gridN    = (N + T - 1) / T;
  const int gridE    = (E + T - 1) / T;
  const int gridEH4  = (int)((EH4 + T - 1) / T);
  const int gridGemm = N / 16;                    // N % 16 == 0

  // degrees -> dinv
  zero_f32x4<<<(N / 4 + T - 1) / T, T, 0, stream>>>((float4*)dinv, N / 4);
  deg_count<<<gridE, T, 0, stream>>>(dst, dinv, E);
  deg_to_dinv<<<gridN, T, 0, stream>>>(dinv, N);

  // ---- layer 1: xw = gather(emb, x_ids) @ W1 ----
  wmma_gemm_f32<128><<<gridGemm, 128, 0, stream>>>(emb, x_ids, W1, bufB);
  zero_f32x4<<<gridNH4, T, 0, stream>>>((float4*)bufA, NH4);
  edge_scatter4<<<gridEH4, T, 0, stream>>>(src, dst, bufB, dinv, bufA, E);
  self_bias_relu4<<<gridNH4, T, 0, stream>>>(bufA, bufB, dinv, b1, N);

  // ---- layer 2 ----
  wmma_gemm_f32<64><<<gridGemm, 128, 0, stream>>>(bufA, nullptr, W2, bufB);
  zero_f32x4<<<gridNH4, T, 0, stream>>>((float4*)bufA, NH4);
  edge_scatter4<<<gridEH4, T, 0, stream>>>(src, dst, bufB, dinv, bufA, E);
  self_bias_relu4<<<gridNH4, T, 0, stream>>>(bufA, bufB, dinv, b2, N);

  // ---- layer 3 ----
  wmma_gemm_f32<64><<<gridGemm, 128, 0, stream>>>(bufA, nullptr, W3, bufB);
  zero_f32x4<<<gridNH4, T, 0, stream>>>((float4*)bufA, NH4);
  edge_scatter4<<<gridEH4, T, 0, stream>>>(src, dst, bufB, dinv, bufA, E);
  self_bias_relu4<<<gridNH4, T, 0, stream>>>(bufA, bufB, dinv, b3, N);

  // ---- global mean pool ----
  zero_f32x4<<<(G * HID / 4 + T - 1) / T, T, 0, stream>>>((float4*)pooled, G * HID / 4);
  zero_f32x4<<<1, T, 0, stream>>>((float4*)counts, G / 4);
  pool_count<<<gridN, T, 0, stream>>>(batch, counts, N);
  pool_sum4<<<gridNH4, T, 0, stream>>>(bufA, batch, pooled, N);
  pool_div<<<(G * HID + T - 1) / T, T, 0, stream>>>(pooled, counts, G);

  // ---- MLP head ----
  mlp_head<<<G, HID, 0, stream>>>(pooled, Wf1, bf1, Wf2, bf2, out);
}